// Nmp_27376121544811
// MI455X (gfx1250) — compile-verified
//
#include <hip/hip_runtime.h>
#include <hip/hip_bf16.h>

#define NN 30000
#define NE 960000
#define F0 64
#define FE 16
#define NOUT 64
#define NTARGET 12
#define MAXDEG 128

typedef __attribute__((ext_vector_type(2))) float v2f;
typedef __attribute__((ext_vector_type(8))) float v8f;

// ---------------- zero fill ----------------
__global__ void k_zero(float* __restrict__ p, int n) {
  int i = blockIdx.x * blockDim.x + threadIdx.x;
  if (i < n) p[i] = 0.f;
}

// ------- edge pass 0: degree + edge-feature aggregation (layer-invariant) -------
// one thread per (edge, 4-feature quad): b128 gather, 4 dword atomics
__global__ void k_edge_efeat_deg(const float* __restrict__ ef,
                                 const int* __restrict__ dst,
                                 float* __restrict__ agg_e,
                                 int* __restrict__ deg) {
  int t = blockIdx.x * blockDim.x + threadIdx.x;
  if (t >= NE * 4) return;                   // 3.84M threads
  int e = t >> 2, q = (t & 3) << 2;
  int d = dst[e];
  const float4 v = *(const float4*)(ef + (size_t)e * FE + q);
  float* p = agg_e + (size_t)d * FE + q;
  atomicAdd(p + 0, v.x);
  atomicAdd(p + 1, v.y);
  atomicAdd(p + 2, v.z);
  atomicAdd(p + 3, v.w);
  if (q == 0) atomicAdd(&deg[d], 1);
}

// ------- edge pass: agg_h[dst] += h[src]  (16 quads/edge, L2-resident atomics) -------
__global__ void k_edge_scatter_h(const float* __restrict__ h,
                                 const int* __restrict__ src,
                                 const int* __restrict__ dst,
                                 float* __restrict__ agg_h) {
  int t = blockIdx.x * blockDim.x + threadIdx.x;
  if (t >= NE * 16) return;                  // 15.36M threads
  int e = t >> 4, q = (t & 15) << 2;
  const float4 v = *(const float4*)(h + (size_t)src[e] * 64 + q);
  float* p = agg_h + (size_t)dst[e] * 64 + q;
  atomicAdd(p + 0, v.x);
  atomicAdd(p + 1, v.y);
  atomicAdd(p + 2, v.z);
  atomicAdd(p + 3, v.w);
}

// ------- node update: h_out = sigmoid([agg_h(64), agg_e(16)] @ H[deg]) , one wave/node -------
__global__ void k_node_update(const float* __restrict__ agg_h,
                              const float* __restrict__ agg_e,
                              const int* __restrict__ deg,
                              const float* __restrict__ H,   // [128,80,64]
                              float* __restrict__ h_out) {
  int gid  = blockIdx.x * blockDim.x + threadIdx.x;
  int node = gid >> 5;                       // wave-uniform
  int lane = threadIdx.x & 31;
  if (node >= NN) return;
  node = __builtin_amdgcn_readfirstlane(node);   // force SGPR -> scalar loads below
  int d = deg[node];
  if (d > MAXDEG - 1) d = MAXDEG - 1;
  const float* Hd = H + (size_t)d * 80 * 64;
  float acc0 = 0.f, acc1 = 0.f;
  const float* mh = agg_h + (size_t)node * 64;
  const float* me = agg_e + (size_t)node * FE;
  #pragma unroll 4
  for (int i = 0; i < 64; ++i) {
    float mi = mh[i];                        // uniform address -> s_load broadcast
    acc0 += mi * Hd[i * 64 + lane];
    acc1 += mi * Hd[i * 64 + lane + 32];
  }
  #pragma unroll 4
  for (int i = 0; i < 16; ++i) {
    float mi = me[i];
    acc0 += mi * Hd[(64 + i) * 64 + lane];
    acc1 += mi * Hd[(64 + i) * 64 + lane + 32];
  }
  h_out[(size_t)node * 64 + lane]      = 1.f / (1.f + __expf(-acc0));
  h_out[(size_t)node * 64 + lane + 32] = 1.f / (1.f + __expf(-acc1));
}

// ------- readout: rvec[64] += sum_rows softmax(h @ W) ; one wave per 16-row tile -------
// Uses V_WMMA_F32_16X16X4_F32: A = 16x4 f32 (v2f/lane), B = 4x16 f32, C/D = 16x16 f32 (v8f).
__global__ void k_readout_softmax(const float* __restrict__ h,   // [NN,64]
                                  const float* __restrict__ W,   // [64,64]
                                  float* __restrict__ rvec) {    // [64]
  const int ntiles = NN / 16;                // 1875 exactly
  int wave = (blockIdx.x * blockDim.x + threadIdx.x) >> 5;   // wave-uniform bound
  int lane = threadIdx.x & 31;
  if (wave >= ntiles) return;                // whole-wave exit -> EXEC all 1s for WMMA
  int m0   = wave * 16;
  int half = lane >> 4;                      // 0: K even pair / rows 0-7 ; 1: K odd pair / rows 8-15
  int l16  = lane & 15;

  v8f c[4] = {};                             // four 16-col accumulator tiles (cols t*16 + l16)
  const float* arow = h + (size_t)(m0 + l16) * 64;
  for (int kc = 0; kc < 16; ++kc) {          // K = 64 in steps of 4
    v2f a;
    a.x = arow[4 * kc + 2 * half + 0];       // A-layout: v0={K0|K2}, v1={K1|K3}
    a.y = arow[4 * kc + 2 * half + 1];
    #pragma unroll
    for (int t = 0; t < 4; ++t) {
      int col = t * 16 + l16;
      v2f b;
      b.x = W[(4 * kc + 2 * half + 0) * 64 + col];
      b.y = W[(4 * kc + 2 * half + 1) * 64 + col];
      c[t] = __builtin_amdgcn_wmma_f32_16x16x4_f32(
          /*neg_a=*/false, a, /*neg_b=*/false, b,
          /*c_mod=*/(short)0, c[t], /*reuse_a=*/false, /*reuse_b=*/false);
    }
  }

  // C layout: vgpr r -> row (r + 8*half), lane l16 -> col within tile.
  // Row-wise softmax: reduce across 4 tiles (regs) x 16 lanes of own half.
  float colsum[4] = {0.f, 0.f, 0.f, 0.f};
  #pragma unroll
  for (int r = 0; r < 8; ++r) {
    float m = fmaxf(fmaxf(c[0][r], c[1][r]), fmaxf(c[2][r], c[3][r]));
    for (int off = 1; off < 16; off <<= 1)
      m = fmaxf(m, __shfl_xor(m, off, 32));  // stays within 16-lane half
    float e0 = __expf(c[0][r] - m), e1 = __expf(c[1][r] - m);
    float e2 = __expf(c[2][r] - m), e3 = __expf(c[3][r] - m);
    float s = e0 + e1 + e2 + e3;
    for (int off = 1; off < 16; off <<= 1)
      s += __shfl_xor(s, off, 32);
    float inv = 1.f / s;
    colsum[0] += e0 * inv; colsum[1] += e1 * inv;
    colsum[2] += e2 * inv; colsum[3] += e3 * inv;
  }
  #pragma unroll
  for (int t = 0; t < 4; ++t) {
    float v = colsum[t] + __shfl_xor(colsum[t], 16, 32);  // rows 0-7 + rows 8-15
    if (half == 0) atomicAdd(&rvec[t * 16 + l16], v);
  }
}

// ------- final linear: out[12] = rvec @ Wr + br -------
__global__ void k_final(const float* __restrict__ rvec,
                        const float* __restrict__ Wr,   // [64,12]
                        const float* __restrict__ br,
                        float* __restrict__ out) {
  int t = threadIdx.x;
  if (t < NTARGET) {
    float acc = br[t];
    #pragma unroll 8
    for (int i = 0; i < 64; ++i) acc += rvec[i] * Wr[i * NTARGET + t];
    out[t] = acc;
  }
}

extern "C" void kernel_launch(void* const* d_in, const int* in_sizes, int n_in,
                              void* d_out, int out_size, void* d_ws, size_t ws_size,
                              hipStream_t stream) {
  const float* h_in   = (const float*)d_in[0];
  const float* e_feat = (const float*)d_in[1];
  const int*   src    = (const int*)d_in[2];
  const int*   dst    = (const int*)d_in[3];
  const float* H0     = (const float*)d_in[4];
  const float* H1     = (const float*)d_in[5];
  const float* W0     = (const float*)d_in[6];
  const float* W1     = (const float*)d_in[7];
  const float* W2     = (const float*)d_in[8];
  const float* Wr     = (const float*)d_in[9];
  const float* br     = (const float*)d_in[10];
  float* out = (float*)d_out;

  // carve workspace (256B aligned regions)
  char* ws = (char*)d_ws;
  size_t off = 0;
  auto carve = [&](size_t bytes) -> char* {
    char* p = ws + off;
    off = (off + bytes + 255) & ~(size_t)255;
    return p;
  };
  int*   deg   = (int*)  carve((size_t)NN * sizeof(int));
  float* agg_e = (float*)carve((size_t)NN * FE * sizeof(float));
  float* agg_h = (float*)carve((size_t)NN * 64 * sizeof(float));
  float* rvec  = (float*)carve(64 * sizeof(float));
  float* h1    = (float*)carve((size_t)NN * 64 * sizeof(float));
  float* h2    = (float*)carve((size_t)NN * 64 * sizeof(float));
  (void)ws_size; (void)in_sizes; (void)n_in; (void)out_size;

  const int B = 256;
  auto blocks = [](long long n, int b) { return (unsigned)((n + b - 1) / b); };

  // fresh accumulators every call (graph-replay safe)
  k_zero<<<blocks(NN, B), B, 0, stream>>>((float*)deg, NN);
  k_zero<<<blocks((long long)NN * FE, B), B, 0, stream>>>(agg_e, NN * FE);
  k_zero<<<blocks((long long)NN * 64, B), B, 0, stream>>>(agg_h, NN * 64);
  k_zero<<<1, 64, 0, stream>>>(rvec, 64);

  // degree + edge-feature aggregation (shared across both MP layers)
  k_edge_efeat_deg<<<blocks((long long)NE * 4, B), B, 0, stream>>>(e_feat, dst, agg_e, deg);

  // layer 1
  k_edge_scatter_h<<<blocks((long long)NE * 16, B), B, 0, stream>>>(h_in, src, dst, agg_h);
  k_node_update<<<blocks((long long)NN * 32, B), B, 0, stream>>>(agg_h, agg_e, deg, H0, h1);

  // layer 2
  k_zero<<<blocks((long long)NN * 64, B), B, 0, stream>>>(agg_h, NN * 64);
  k_edge_scatter_h<<<blocks((long long)NE * 16, B), B, 0, stream>>>(h1, src, dst, agg_h);
  k_node_update<<<blocks((long long)NN * 32, B), B, 0, stream>>>(agg_h, agg_e, deg, H1, h2);

  // readout (WMMA f32 GEMM + row softmax + node-sum), accumulated into rvec
  unsigned rblocks = blocks((long long)(NN / 16) * 32, B);
  k_readout_softmax<<<rblocks, B, 0, stream>>>(h_in, W0, rvec);
  k_readout_softmax<<<rblocks, B, 0, stream>>>(h1,   W1, rvec);
  k_readout_softmax<<<rblocks, B, 0, stream>>>(h2,   W2, rvec);

  // final 64 -> 12 linear
  k_final<<<1, 16, 0, stream>>>(rvec, Wr, br, out);
}